// LDRFLIPLoss_33612414058812
// MI455X (gfx1250) — compile-verified
//
#include <hip/hip_runtime.h>
#include <math.h>

// ---------------------------------------------------------------------------
// LDR-FLIP on gfx1250 (MI455X).  f32 separable convolutions expressed as
// banded-matrix GEMMs on V_WMMA_F32_16X16X4_F32; wave32 owns a 16x16 tile.
// All 1D filters unified to 21 taps (R=10, feature kernels zero-padded) so
// KEXT = 36 = 9 K-chunks exactly; weights live in a zero-guarded LDS band and
// are preloaded to registers; interior tiles use immediate-offset loads.
// ---------------------------------------------------------------------------

typedef float v2f __attribute__((ext_vector_type(2)));
typedef float v8f __attribute__((ext_vector_type(8)));

#define WMMA_F32X4(a, b, c) \
  __builtin_amdgcn_wmma_f32_16x16x4_f32(false, (a), false, (b), (short)0, (c), false, false)

// ---------------- problem geometry (fixed by the reference) ----------------
constexpr int W_       = 1920;
constexpr int Himg_    = 1080;
constexpr int B_       = 2;
constexpr int HW_      = W_ * Himg_;        // 2,073,600
constexpr int P_       = B_ * HW_;          // 4,147,200  (one batched plane)
constexpr int RowsTot_ = B_ * Himg_;        // 2160 (multiple of 16)
constexpr int R_       = 10;                // unified radius
constexpr int WPAD     = 56;                // 16 zeros | 21 taps | 19 zeros

constexpr int TILES_X   = W_ / 16;          // 120
constexpr int TILES_YH  = RowsTot_ / 16;    // 135 (horizontal pass, batched rows)
constexpr int TILES_YV  = (Himg_ + 15)/16;  // 68  (vertical pass, per image)

// ---------------- filter constants -----------------------------------------
constexpr double PPD_D   = 3840.0 * 3.14159265358979323846 / 180.0; // 67.0206
constexpr double PI2_D   = 9.86960440108935861883;
constexpr double A2C     = PI2_D / (PPD_D * PPD_D);                 // pi^2*dx^2
constexpr float  ALPHA_A   = (float)(A2C / 0.0047);
constexpr float  ALPHA_RG  = (float)(A2C / 0.0053);
constexpr float  ALPHA_B1  = (float)(A2C / 0.04);
constexpr float  ALPHA_B2  = (float)(A2C / 0.025);
constexpr float  SD_F      = (float)(0.5 * 0.082 * PPD_D);          // 2.74785
constexpr float  EPS_      = 1e-15f;

// ---------------- workspace layout (floats) ---------------------------------
// [0,3P)   yc_t (Y,Cx,Cz)  -> later overwritten by pre_t (L, 0.01La, 0.01Lb)
// [3P,6P)  yc_r            -> later pre_r
// [6P,7P)  phi_t
// [7P,8P)  phi_r
// [8P,16P) CSF horizontal planes (4 per input); reused for feature h-planes
// [16P,17P) de_c
// [17P,..) block partial sums (2*2040 floats)
#define OFF_YC_T  ((size_t)0)
#define OFF_YC_R  ((size_t)3 * P_)
#define OFF_PHI_T ((size_t)6 * P_)
#define OFF_PHI_R ((size_t)7 * P_)
#define OFF_H     ((size_t)8 * P_)
#define OFF_DEC   ((size_t)16 * P_)
#define OFF_PART  ((size_t)17 * P_)

// ---------------- color helpers ---------------------------------------------
__device__ __forceinline__ float srgb2lin(float c) {
  return (c > 0.04045f) ? __powf((c + 0.055f) / 1.055f, 2.4f) : c * (1.0f / 12.92f);
}

__device__ __forceinline__ void mat_l2x(float r, float g, float b,
                                        float& X, float& Y, float& Z) {
  X = (10135552.0f/24577794.0f)*r + (8788810.0f/24577794.0f)*g + (4435075.0f/24577794.0f)*b;
  Y = (2613072.0f/12288897.0f)*r + (8788810.0f/12288897.0f)*g + (887015.0f/12288897.0f)*b;
  Z = (1425312.0f/73733382.0f)*r + (8788810.0f/73733382.0f)*g + (70074185.0f/73733382.0f)*b;
}

// hunt(xyz2lab(A_L2X * rgb)) for a linear-RGB triple
__device__ __forceinline__ void hunt_from_rgb(float r, float g, float b, float* o) {
  float X, Y, Z;
  mat_l2x(r, g, b, X, Y, Z);
  X *= 1.052156925f;  /* Y *= 1 */  Z *= 0.91835767f;
  const float d3 = (6.0f/29.0f)*(6.0f/29.0f)*(6.0f/29.0f);
  float fx = (X > d3) ? cbrtf(X) : X * (841.0f/108.0f) + (4.0f/29.0f);
  float fy = (Y > d3) ? cbrtf(Y) : Y * (841.0f/108.0f) + (4.0f/29.0f);
  float fz = (Z > d3) ? cbrtf(Z) : Z * (841.0f/108.0f) + (4.0f/29.0f);
  float L  = 116.0f * fy - 16.0f;
  float a  = 500.0f * (fx - fy);
  float bb = 200.0f * (fy - fz);
  o[0] = L; o[1] = 0.01f * L * a; o[2] = 0.01f * L * bb;
}

__device__ __forceinline__ float hyab3(const float* p, const float* q) {
  float dL = p[0]-q[0], da = p[1]-q[1], db = p[2]-q[2];
  return sqrtf(fmaxf(dL*dL, EPS_)) + sqrtf(da*da + db*db);
}

// ---------------- weight builders (thread 0, into zero-guarded LDS bands) ---
// Band layout: wp[16+t] = tap t (21 taps), zeros elsewhere.
__device__ __forceinline__ void fill_gauss21(float* tap0, float alpha, bool norm) {
  float s = 0.0f;
  for (int t = 0; t < 21; ++t) {
    float x = (float)(t - 10);
    tap0[t] = expf(-alpha * x * x);
    s += tap0[t];
  }
  if (norm) { float inv = 1.0f / s; for (int t = 0; t < 21; ++t) tap0[t] *= inv; }
}

// 19-tap feature kernels written centered (offset +1 inside the 21-tap band)
__device__ __forceinline__ void fill_feat(float* gh0, float* fe0, float* fp0) {
  const float sd = SD_F;
  float g[19], fer[19], fpr[19];
  float Sg = 0.f, Ne = 0.f, Pe = 0.f, Np = 0.f, Pp = 0.f;
  for (int t = 0; t < 19; ++t) {
    float x = (float)(t - 9);
    g[t] = expf(-(x * x) / (2.0f * sd * sd));
    Sg += g[t];
    fer[t] = -x * g[t];
    fpr[t] = (x * x / (sd * sd) - 1.0f) * g[t];
    if (fer[t] < 0.f) Ne -= fer[t]; else Pe += fer[t];
    if (fpr[t] < 0.f) Np -= fpr[t]; else Pp += fpr[t];
  }
  for (int t = 0; t < 19; ++t) {
    gh0[t + 1] = g[t] / Sg;
    fe0[t + 1] = (fer[t] < 0.f) ? fer[t] / Ne : fer[t] / Pe;
    fp0[t + 1] = (fpr[t] < 0.f) ? fpr[t] / Np : fpr[t] / Pp;
  }
}

__device__ __forceinline__ void zero_bands(float* base, int nfloats) {
  for (int i = threadIdx.x; i < nfloats; i += blockDim.x) base[i] = 0.0f;
}

// ---------------- banded-weight preload (LDS -> 9 x v2f registers) ----------
// Per lane: band index base = 16 + 2*hi - m; chunk c uses [base+4c, base+4c+1].
// Serves as the B operand of the horizontal pass and the A operand of the
// vertical pass (both are w[d] with d = (4c + 2*hi) - lane_row).
__device__ __forceinline__ void load_wband(const float* __restrict__ wp, int lane,
                                           v2f wreg[9]) {
  const int base = 16 + 2 * (lane >> 4) - (lane & 15);
#pragma unroll
  for (int c = 0; c < 9; ++c) {
    wreg[c].x = wp[base + 4 * c];
    wreg[c].y = wp[base + 4 * c + 1];
  }
}

// ---------------- WMMA tile convolutions ------------------------------------
// Horizontal, interior tile: pure immediate-offset b64 loads.
__device__ __forceinline__ v8f hconv_int(const float* __restrict__ plane, int i0, int j0,
                                         const v2f wreg[9], int lane) {
  const int m = lane & 15, hi = lane >> 4;
  const float* rp = plane + (size_t)(i0 + m) * W_ + (j0 - R_ + 2 * hi);
  v8f acc = {};
#pragma unroll
  for (int c = 0; c < 9; ++c) {
    v2f a = *(const v2f*)(rp + 4 * c);       // 8B-aligned: j0,R_,2hi all even
    acc = WMMA_F32X4(a, wreg[c], acc);
  }
  return acc;
}

// Horizontal, boundary tile: clamped columns ('edge' padding).
__device__ __forceinline__ v8f hconv_edge(const float* __restrict__ plane, int i0, int j0,
                                          const v2f wreg[9], int lane) {
  const int m = lane & 15, hi = lane >> 4;
  const float* rp = plane + (size_t)(i0 + m) * W_;
  v8f acc = {};
#pragma unroll
  for (int c = 0; c < 9; ++c) {
    int c0 = j0 - R_ + 4 * c + 2 * hi;
    int c1 = c0 + 1;
    c0 = c0 < 0 ? 0 : (c0 > W_ - 1 ? W_ - 1 : c0);
    c1 = c1 < 0 ? 0 : (c1 > W_ - 1 ? W_ - 1 : c1);
    v2f a; a.x = rp[c0]; a.y = rp[c1];
    acc = WMMA_F32X4(a, wreg[c], acc);
  }
  return acc;
}

__device__ __forceinline__ v8f hconv(const float* plane, int i0, int j0,
                                     const v2f wreg[9], int lane) {
  if (j0 >= R_ && j0 + 15 + R_ < W_) return hconv_int(plane, i0, j0, wreg, lane);
  return hconv_edge(plane, i0, j0, wreg, lane);
}

// Vertical, interior tile: per-chunk row offsets are compile-time immediates.
__device__ __forceinline__ v8f vconv_int(const float* __restrict__ plane, int img,
                                         int i0, int j0, const v2f wreg[9], int lane) {
  const int m = lane & 15, hi = lane >> 4;
  const float* cp = plane + (size_t)img * HW_ + (size_t)(i0 - R_) * W_ + j0 + m;
  v8f acc = {};
#pragma unroll
  for (int c = 0; c < 9; ++c) {
    const int k0 = 4 * c + 2 * hi;
    v2f b;
    b.x = cp[(size_t)k0 * W_];
    b.y = cp[(size_t)(k0 + 1) * W_];
    acc = WMMA_F32X4(wreg[c], b, acc);
  }
  return acc;
}

// Vertical, boundary tile: clamped rows ('edge' padding).
__device__ __forceinline__ v8f vconv_edge(const float* __restrict__ plane, int img,
                                          int i0, int j0, const v2f wreg[9], int lane) {
  const int m = lane & 15, hi = lane >> 4;
  const float* base = plane + (size_t)img * HW_ + j0 + m;
  v8f acc = {};
#pragma unroll
  for (int c = 0; c < 9; ++c) {
    int r0 = i0 - R_ + 4 * c + 2 * hi;
    int r1 = r0 + 1;
    r0 = r0 < 0 ? 0 : (r0 > Himg_ - 1 ? Himg_ - 1 : r0);
    r1 = r1 < 0 ? 0 : (r1 > Himg_ - 1 ? Himg_ - 1 : r1);
    v2f b;
    b.x = base[(size_t)r0 * W_];
    b.y = base[(size_t)r1 * W_];
    acc = WMMA_F32X4(wreg[c], b, acc);
  }
  return acc;
}

__device__ __forceinline__ v8f vconv(const float* plane, int img, int i0, int j0,
                                     const v2f wreg[9], int lane) {
  if (i0 >= R_ && i0 + 15 + R_ < Himg_) return vconv_int(plane, img, i0, j0, wreg, lane);
  return vconv_edge(plane, img, i0, j0, wreg, lane);
}

// ---------------- kernel 1: pointwise sRGB -> YCxCz + phi --------------------
__global__ void k_pre(const float* __restrict__ test, const float* __restrict__ target,
                      float* __restrict__ ws) {
  size_t p = (size_t)blockIdx.x * blockDim.x + threadIdx.x;
  if (p >= (size_t)P_) return;
  size_t b = p / HW_, hw = p % HW_;
  const float* srcs[2] = {test, target};
#pragma unroll
  for (int which = 0; which < 2; ++which) {
    const float* src = srcs[which];
    float r = srgb2lin(src[(b * 3 + 0) * HW_ + hw]);
    float g = srgb2lin(src[(b * 3 + 1) * HW_ + hw]);
    float bl = srgb2lin(src[(b * 3 + 2) * HW_ + hw]);
    float X, Y, Z;
    mat_l2x(r, g, bl, X, Y, Z);
    X *= 1.052156925f;  Z *= 0.91835767f;      // INV_REF_ILLUM (Y*1)
    float y  = 116.0f * Y - 16.0f;
    float cx = 500.0f * (X - Y);
    float cz = 200.0f * (Y - Z);
    float* yc = ws + (which == 0 ? OFF_YC_T : OFF_YC_R);
    yc[p] = y; yc[(size_t)P_ + p] = cx; yc[2 * (size_t)P_ + p] = cz;
    ws[(which == 0 ? OFF_PHI_T : OFF_PHI_R) + p] = (y + 16.0f) / 116.0f;
  }
}

// ---------------- kernel 2: horizontal CSF (8 planes) ------------------------
// z: 0..7 = {A(Y), RG(Cx), BY1(Cz), BY2(Cz)} x {test, ref}
__global__ __launch_bounds__(128) void k_hcsf(float* __restrict__ ws) {
  __shared__ float wp[4][WPAD];
  zero_bands(&wp[0][0], 4 * WPAD);
  __syncthreads();
  if (threadIdx.x == 0) {
    fill_gauss21(&wp[0][16], ALPHA_A, true);
    fill_gauss21(&wp[1][16], ALPHA_RG, true);
    fill_gauss21(&wp[2][16], ALPHA_B1, false);   // raw; normalization in vpass
    fill_gauss21(&wp[3][16], ALPHA_B2, false);
  }
  __syncthreads();
  const int wave = threadIdx.x >> 5, lane = threadIdx.x & 31;
  const int tile = blockIdx.x * 4 + wave;               // 0..16199
  const int z = blockIdx.y;
  const int i0 = (tile / TILES_X) * 16, j0 = (tile % TILES_X) * 16;
  const int kind = z & 3;
  const int ch = (kind == 0) ? 0 : (kind == 1) ? 1 : 2;
  const float* in = ws + (z < 4 ? OFF_YC_T : OFF_YC_R) + (size_t)ch * P_;
  float* out = ws + OFF_H + (size_t)z * P_;
  v2f wreg[9];
  load_wband(wp[kind], lane, wreg);
  v8f acc = hconv(in, i0, j0, wreg, lane);
  const int m = lane & 15, hi = lane >> 4;
#pragma unroll
  for (int v = 0; v < 8; ++v)
    out[(size_t)(i0 + v + 8 * hi) * W_ + j0 + m] = acc[v];
}

// ---------------- kernel 3: vertical CSF + color pipeline + Hunt -------------
// z: 0..3 = {test,ref} x {image 0,1}
__global__ __launch_bounds__(128) void k_vcsf(float* __restrict__ ws) {
  __shared__ float wp[4][WPAD];
  zero_bands(&wp[0][0], 4 * WPAD);
  __syncthreads();
  if (threadIdx.x == 0) {
    fill_gauss21(&wp[0][16], ALPHA_A, true);
    fill_gauss21(&wp[1][16], ALPHA_RG, true);
    float u1[21], u2[21];
    float S1 = 0.f, S2 = 0.f;
    for (int t = 0; t < 21; ++t) {
      float x = (float)(t - 10);
      u1[t] = expf(-ALPHA_B1 * x * x); S1 += u1[t];
      u2[t] = expf(-ALPHA_B2 * x * x); S2 += u2[t];
    }
    float c1 = 34.1f * sqrtf((float)(3.14159265358979323846 / 0.04));
    float c2 = 13.5f * sqrtf((float)(3.14159265358979323846 / 0.025));
    float inv = 1.0f / (c1 * S1 * S1 + c2 * S2 * S2);
    for (int t = 0; t < 21; ++t) {
      wp[2][16 + t] = c1 * u1[t] * inv;
      wp[3][16 + t] = c2 * u2[t] * inv;
    }
  }
  __syncthreads();
  const int wave = threadIdx.x >> 5, lane = threadIdx.x & 31;
  const int tile = blockIdx.x * 4 + wave;               // 0..8159
  const int z = blockIdx.y;
  const int inp = z >> 1, img = z & 1;
  const int i0 = (tile / TILES_X) * 16, j0 = (tile % TILES_X) * 16;
  const float* hb = ws + OFF_H + (size_t)inp * 4 * P_;
  float* pre = ws + (size_t)inp * 3 * P_;               // overwrite yc planes
  v2f wA[9], wRG[9], wB1[9], wB2[9];
  load_wband(wp[0], lane, wA);
  load_wband(wp[1], lane, wRG);
  load_wband(wp[2], lane, wB1);
  load_wband(wp[3], lane, wB2);
  v8f Yf  = vconv(hb + 0 * (size_t)P_, img, i0, j0, wA,  lane);
  v8f Cxf = vconv(hb + 1 * (size_t)P_, img, i0, j0, wRG, lane);
  v8f Cz1 = vconv(hb + 2 * (size_t)P_, img, i0, j0, wB1, lane);
  v8f Cz2 = vconv(hb + 3 * (size_t)P_, img, i0, j0, wB2, lane);
  const int m = lane & 15, hi = lane >> 4;
#pragma unroll
  for (int v = 0; v < 8; ++v) {
    const int row = i0 + v + 8 * hi;
    if (row < Himg_) {
      float Y = Yf[v], Cx = Cxf[v], Cz = Cz1[v] + Cz2[v];
      // ycxcz -> xyz
      float yy = (Y + 16.0f) / 116.0f;
      float X = (yy + Cx * (1.0f/500.0f)) * 0.950428545f;
      float Yx = yy;
      float Z = (yy - Cz * (1.0f/200.0f)) * 1.088900371f;
      // xyz -> linear rgb, clip [0,1]
      float R  =  3.241003275f * X - 1.537398934f * Yx - 0.498615861f * Z;
      float G  = -0.969224334f * X + 1.875930071f * Yx + 0.041554224f * Z;
      float Bv =  0.055639423f * X - 0.204011202f * Yx + 1.057148933f * Z;
      R = fminf(fmaxf(R, 0.f), 1.f); G = fminf(fmaxf(G, 0.f), 1.f); Bv = fminf(fmaxf(Bv, 0.f), 1.f);
      float o[3];
      hunt_from_rgb(R, G, Bv, o);
      size_t px = (size_t)img * HW_ + (size_t)row * W_ + j0 + m;
      pre[px] = o[0]; pre[(size_t)P_ + px] = o[1]; pre[2 * (size_t)P_ + px] = o[2];
    }
  }
}

// ---------------- kernel 4: chromatic error de_c -----------------------------
__global__ void k_dec(float* __restrict__ ws) {
  size_t p = (size_t)blockIdx.x * blockDim.x + threadIdx.x;
  if (p >= (size_t)P_) return;
  const float* pt = ws + OFF_YC_T;   // pre_t
  const float* pr = ws + OFF_YC_R;   // pre_r
  float a[3] = {pt[p], pt[(size_t)P_ + p], pt[2 * (size_t)P_ + p]};
  float b[3] = {pr[p], pr[(size_t)P_ + p], pr[2 * (size_t)P_ + p]};
  float de = __powf(hyab3(b, a), 0.7f);
  float gh[3], bh[3];
  hunt_from_rgb(0.f, 1.f, 0.f, gh);
  hunt_from_rgb(0.f, 0.f, 1.f, bh);
  float cmax = __powf(hyab3(gh, bh), 0.7f);
  float pccmax = 0.4f * cmax;
  float dec = (de < pccmax) ? de * (0.95f / pccmax)
                            : 0.95f + (de - pccmax) / (cmax - pccmax) * 0.05f;
  ws[OFF_DEC + p] = dec;
}

// ---------------- kernel 5: horizontal feature filters -----------------------
// z: 0..5 = {edge_x, point_x, gauss_x} x {test, ref}   on phi planes
__global__ __launch_bounds__(128) void k_hfeat(float* __restrict__ ws) {
  __shared__ float wp[3][WPAD];  // gh, fe, fp (21-tap bands, ends zero)
  zero_bands(&wp[0][0], 3 * WPAD);
  __syncthreads();
  if (threadIdx.x == 0) fill_feat(&wp[0][16], &wp[1][16], &wp[2][16]);
  __syncthreads();
  const int wave = threadIdx.x >> 5, lane = threadIdx.x & 31;
  const int tile = blockIdx.x * 4 + wave;
  const int z = blockIdx.y;
  const int i0 = (tile / TILES_X) * 16, j0 = (tile % TILES_X) * 16;
  const int kind = z % 3;  // 0 edge, 1 point, 2 gauss
  const float* wsel = (kind == 0) ? wp[1] : (kind == 1) ? wp[2] : wp[0];
  const float* in = ws + (z < 3 ? OFF_PHI_T : OFF_PHI_R);
  float* out = ws + OFF_H + (size_t)z * P_;
  v2f wreg[9];
  load_wband(wsel, lane, wreg);
  v8f acc = hconv(in, i0, j0, wreg, lane);
  const int m = lane & 15, hi = lane >> 4;
#pragma unroll
  for (int v = 0; v < 8; ++v)
    out[(size_t)(i0 + v + 8 * hi) * W_ + j0 + m] = acc[v];
}

// ---------------- kernel 6: vertical features + de_f + combine + reduce ------
__global__ __launch_bounds__(128) void k_vfeat(float* __restrict__ ws,
                                               float* __restrict__ partial) {
  __shared__ float wp[3][WPAD];  // gh, fe, fp
  __shared__ float red[128];
  zero_bands(&wp[0][0], 3 * WPAD);
  __syncthreads();
  if (threadIdx.x == 0) fill_feat(&wp[0][16], &wp[1][16], &wp[2][16]);
  __syncthreads();
  const int wave = threadIdx.x >> 5, lane = threadIdx.x & 31;
  const int tile = blockIdx.x * 4 + wave;               // 0..8159
  const int img = blockIdx.y;
  const int i0 = (tile / TILES_X) * 16, j0 = (tile % TILES_X) * 16;
  const float* fh = ws + OFF_H;   // he_t, hp_t, hg_t, he_r, hp_r, hg_r
  v2f wG[9], wE[9], wPp[9];
  load_wband(wp[0], lane, wG);
  load_wband(wp[1], lane, wE);
  load_wband(wp[2], lane, wPp);
  // test features
  v8f fxT = vconv(fh + 0 * (size_t)P_, img, i0, j0, wG,  lane);
  v8f fyT = vconv(fh + 2 * (size_t)P_, img, i0, j0, wE,  lane);
  v8f pxT = vconv(fh + 1 * (size_t)P_, img, i0, j0, wG,  lane);
  v8f pyT = vconv(fh + 2 * (size_t)P_, img, i0, j0, wPp, lane);
  // ref features
  v8f fxR = vconv(fh + 3 * (size_t)P_, img, i0, j0, wG,  lane);
  v8f fyR = vconv(fh + 5 * (size_t)P_, img, i0, j0, wE,  lane);
  v8f pxR = vconv(fh + 4 * (size_t)P_, img, i0, j0, wG,  lane);
  v8f pyR = vconv(fh + 5 * (size_t)P_, img, i0, j0, wPp, lane);
  const float* dec = ws + OFF_DEC;
  const int m = lane & 15, hi = lane >> 4;
  float lsum = 0.0f;
#pragma unroll
  for (int v = 0; v < 8; ++v) {
    const int row = i0 + v + 8 * hi;
    if (row < Himg_) {
      float nET = sqrtf(fxT[v]*fxT[v] + fyT[v]*fyT[v]);
      float nPT = sqrtf(pxT[v]*pxT[v] + pyT[v]*pyT[v]);
      float nER = sqrtf(fxR[v]*fxR[v] + fyR[v]*fyR[v]);
      float nPR = sqrtf(pxR[v]*pxR[v] + pyR[v]*pyR[v]);
      float def_raw = fmaxf(fabsf(nER - nET), fabsf(nPT - nPR));
      float def = sqrtf(0.70710678118f * def_raw);       // (x/sqrt2)^0.5
      size_t px = (size_t)img * HW_ + (size_t)row * W_ + j0 + m;
      lsum += __powf(dec[px], 1.0f - def);
    }
  }
  red[threadIdx.x] = lsum;
  __syncthreads();
  for (int s = 64; s > 0; s >>= 1) {
    if (threadIdx.x < s) red[threadIdx.x] += red[threadIdx.x + s];
    __syncthreads();
  }
  if (threadIdx.x == 0)
    partial[(size_t)blockIdx.y * gridDim.x + blockIdx.x] = red[0];
}

// ---------------- kernel 7: deterministic final reduction --------------------
__global__ void k_reduce(const float* __restrict__ partial, int n, float* __restrict__ out) {
  __shared__ float red[256];
  float s = 0.0f;
  for (int i = threadIdx.x; i < n; i += 256) s += partial[i];
  red[threadIdx.x] = s;
  __syncthreads();
  for (int st = 128; st > 0; st >>= 1) {
    if (threadIdx.x < st) red[threadIdx.x] += red[threadIdx.x + st];
    __syncthreads();
  }
  if (threadIdx.x == 0) out[0] = red[0] / (float)P_;
}

// ---------------------------------------------------------------------------
extern "C" void kernel_launch(void* const* d_in, const int* in_sizes, int n_in,
                              void* d_out, int out_size, void* d_ws, size_t ws_size,
                              hipStream_t stream) {
  const float* test   = (const float*)d_in[0];
  const float* target = (const float*)d_in[1];
  float* ws = (float*)d_ws;   // needs ~17*P_ + 4096 floats (~282 MB)
  float* partials = ws + OFF_PART;

  const int pwBlocks = (P_ + 255) / 256;
  const int hBlocks  = TILES_X * TILES_YH / 4;  // 4050 (4 waves/block, 1 tile/wave)
  const int vBlocks  = TILES_X * TILES_YV / 4;  // 2040

  k_pre  <<<pwBlocks, 256, 0, stream>>>(test, target, ws);
  k_hcsf <<<dim3(hBlocks, 8), 128, 0, stream>>>(ws);
  k_vcsf <<<dim3(vBlocks, 4), 128, 0, stream>>>(ws);
  k_dec  <<<pwBlocks, 256, 0, stream>>>(ws);
  k_hfeat<<<dim3(hBlocks, 6), 128, 0, stream>>>(ws);
  k_vfeat<<<dim3(vBlocks, 2), 128, 0, stream>>>(ws, partials);
  k_reduce<<<1, 256, 0, stream>>>(partials, 2 * vBlocks, (float*)d_out);
}